// ModelNew_25812753449962
// MI455X (gfx1250) — compile-verified
//
#include <hip/hip_runtime.h>
#include <hip/hip_bf16.h>
#include <math.h>

typedef __attribute__((ext_vector_type(2))) float v2f;
typedef __attribute__((ext_vector_type(8))) float v8f;

// Problem constants (from the reference)
constexpr int BATCHc = 2;
constexpr int Tc     = 4096;
constexpr int Hc     = 64;
constexpr int Pc     = 64;    // d_head
constexpr int Nc     = 128;   // d_state

// Parallel split of the time axis (final state is a pure sum over t)
constexpr int SPLIT  = 4;
constexpr int TS     = Tc / SPLIT;     // 1024 timesteps per split
constexpr int STG    = 32;             // timesteps per pipeline stage
constexpr int NST    = TS / STG;       // 32 stages per split
constexpr int LPT    = TS / 256;       // A elements per thread in the scan

// Per-stage async copies per thread: X 512/256=2, B 1024/256=4  -> 6 total
// (used as the s_wait_asynccnt threshold while one stage is still in flight)

__device__ __forceinline__ void async_copy_b128(unsigned lds_off, const void* gptr) {
    unsigned long long ga = (unsigned long long)(uintptr_t)gptr;
    asm volatile("global_load_async_to_lds_b128 %0, %1, off"
                 :: "v"(lds_off), "v"(ga) : "memory");
}
__device__ __forceinline__ void async_wait_0() {
    asm volatile("s_wait_asynccnt 0x0" ::: "memory");
}
__device__ __forceinline__ void async_wait_6() {   // allow next stage in flight
    asm volatile("s_wait_asynccnt 0x6" ::: "memory");
}

// Issue the async global->LDS copies for one 32-timestep stage.
__device__ __forceinline__ void stage_issue(
    const float* __restrict__ Xg, const float* __restrict__ Bg,
    int b, int h, int tbase, unsigned xoff, unsigned boff, int tid)
{
    // X: 32t x 64p  = 512 b128 -> 2 per thread
#pragma unroll
    for (int it = 0; it < 2; ++it) {
        const int i  = tid + it * 256;
        const int tl = i >> 4;                  // 16 float4 per row
        const int p4 = (i & 15) << 2;
        async_copy_b128(xoff + (unsigned)(tl * Pc + p4) * 4u,
                        Xg + (((size_t)b * Tc + tbase + tl) * Hc + h) * Pc + p4);
    }
    // B: 32t x 128n = 1024 b128 -> 4 per thread
#pragma unroll
    for (int it = 0; it < 4; ++it) {
        const int i  = tid + it * 256;
        const int tl = i >> 5;                  // 32 float4 per row
        const int n4 = (i & 31) << 2;
        async_copy_b128(boff + (unsigned)(tl * Nc + n4) * 4u,
                        Bg + (((size_t)b * Tc + tbase + tl) * Hc + h) * Nc + n4);
    }
}

// -------------------------------------------------------------------------
// Partial kernel: one workgroup per (b, h, split).
// part[p,n] = sum_{t in split} exp(suffix_A(t)) * X[t,p] * B[t,n]
// Double-buffered async->LDS pipeline feeding WMMA f32 16x16x4.
// -------------------------------------------------------------------------
__global__ __launch_bounds__(256) void mamba_final_state_partial(
    const float* __restrict__ Xg, const float* __restrict__ Ag,
    const float* __restrict__ Bg, float* __restrict__ out, int useAtomic)
{
    __shared__ float sW[TS];               // decay weights (4 KB)
    __shared__ float sX[2 * STG * Pc];     // X tiles, double-buffered (16 KB)
    __shared__ float sB[2 * STG * Nc];     // B tiles, double-buffered (32 KB)
    __shared__ float sScan[256];
    __shared__ float sExcl[256];
    __shared__ float sTail;
    __shared__ float sTotal;

    const int tid = threadIdx.x;
    const int bh  = blockIdx.x % (BATCHc * Hc);
    const int s   = blockIdx.x / (BATCHc * Hc);
    const int b   = bh / Hc;
    const int h   = bh % Hc;
    const int t0  = s * TS;

    const float* Aslice = Ag + (size_t)b * Tc * Hc + h;   // stride Hc in t

    // ---- Phase 1a: tail = sum of A over [t0+TS, T) -----------------------
    float tacc = 0.f;
    for (int t = t0 + TS + tid; t < Tc; t += 256) tacc += Aslice[(size_t)t * Hc];
    sScan[tid] = tacc;
    __syncthreads();
    if (tid == 0) {
        float tot = 0.f;
        for (int i = 0; i < 256; ++i) tot += sScan[i];
        sTail = tot;
    }
    __syncthreads();
    const float tail = sTail;

    // ---- Phase 1b: inclusive cumsum of A over the split range ------------
    float av[LPT];
    float run = 0.f;
    for (int j = 0; j < LPT; ++j) {
        av[j] = Aslice[(size_t)(t0 + tid * LPT + j) * Hc];
        run += av[j];
        av[j] = run;
    }
    sScan[tid] = run;                         // safe: prior barrier done
    __syncthreads();
    if (tid == 0) {
        float r = 0.f;
        for (int i = 0; i < 256; ++i) { sExcl[i] = r; r += sScan[i]; }
        sTotal = r;
    }
    __syncthreads();
    {
        const float base = sExcl[tid];
        const float rtot = sTotal;
        for (int j = 0; j < LPT; ++j) {
            const float cum = base + av[j];
            sW[tid * LPT + j] = expf(rtot - cum + tail);   // suffix decay <= 1
        }
    }
    __syncthreads();

    // ---- Phase 2: double-buffered async pipeline + WMMA ------------------
    const int wv   = tid >> 5;          // wave 0..7
    const int lane = tid & 31;
    const int pb   = wv >> 1;           // p-block 0..3  (16 rows each)
    const int ng   = wv & 1;            // n-half  0..1  (64 cols each)
    const int m    = lane & 15;
    const int kb   = (lane < 16) ? 0 : 2;  // A/B frag: lanes 16-31 hold K+2,K+3

    const unsigned xoff0 = (unsigned)(uintptr_t)(&sX[0]);
    const unsigned boff0 = (unsigned)(uintptr_t)(&sB[0]);
    const unsigned xoff1 = xoff0 + (unsigned)(STG * Pc) * 4u;
    const unsigned boff1 = boff0 + (unsigned)(STG * Nc) * 4u;

    v8f acc[4] = {};   // 4 16x16 output tiles per wave

    // Prologue: stage 0 into buffer 0
    stage_issue(Xg, Bg, b, h, t0, xoff0, boff0, tid);

    int buf = 0;
    for (int st = 0; st < NST - 1; ++st) {
        // Issue next stage into the other buffer (it was last read 2 stages
        // ago; the trailing barrier of that compute already passed).
        stage_issue(Xg, Bg, b, h, t0 + (st + 1) * STG,
                    buf ? xoff0 : xoff1, buf ? boff0 : boff1, tid);
        async_wait_6();            // previous 6 copies (stage st) have landed
        __syncthreads();           // visible to all waves

        const float* cX = buf ? (sX + STG * Pc) : sX;
        const float* cB = buf ? (sB + STG * Nc) : sB;
        const int    wb = st * STG;
        for (int kk = 0; kk < STG; kk += 4) {
            const float w0 = sW[wb + kk + kb];
            const float w1 = sW[wb + kk + kb + 1];
            v2f afrag;
            afrag.x = cX[(kk + kb) * Pc + pb * 16 + m] * w0;
            afrag.y = cX[(kk + kb + 1) * Pc + pb * 16 + m] * w1;
#pragma unroll
            for (int q = 0; q < 4; ++q) {
                const int n = ng * 64 + q * 16 + m;
                v2f bfrag;
                bfrag.x = cB[(kk + kb) * Nc + n];
                bfrag.y = cB[(kk + kb + 1) * Nc + n];
                acc[q] = __builtin_amdgcn_wmma_f32_16x16x4_f32(
                    false, afrag, false, bfrag, (short)0, acc[q], false, false);
            }
        }
        __syncthreads();           // done reading before this buffer is reused
        buf ^= 1;
    }

    // Epilogue stage: nothing left to issue; drain fully.
    {
        async_wait_0();
        __syncthreads();
        const float* cX = buf ? (sX + STG * Pc) : sX;
        const float* cB = buf ? (sB + STG * Nc) : sB;
        const int    wb = (NST - 1) * STG;
        for (int kk = 0; kk < STG; kk += 4) {
            const float w0 = sW[wb + kk + kb];
            const float w1 = sW[wb + kk + kb + 1];
            v2f afrag;
            afrag.x = cX[(kk + kb) * Pc + pb * 16 + m] * w0;
            afrag.y = cX[(kk + kb + 1) * Pc + pb * 16 + m] * w1;
#pragma unroll
            for (int q = 0; q < 4; ++q) {
                const int n = ng * 64 + q * 16 + m;
                v2f bfrag;
                bfrag.x = cB[(kk + kb) * Nc + n];
                bfrag.y = cB[(kk + kb + 1) * Nc + n];
                acc[q] = __builtin_amdgcn_wmma_f32_16x16x4_f32(
                    false, afrag, false, bfrag, (short)0, acc[q], false, false);
            }
        }
    }

    // ---- Epilogue: C/D layout -> (b,h,p,n) -------------------------------
    const int nlo = lane & 15;
    const int mhi = (lane >> 4) * 8;
    float* outBase = useAtomic
        ? out + ((size_t)b * Hc + h) * Pc * Nc
        : out + ((size_t)s * BATCHc * Hc + (size_t)b * Hc + h) * Pc * Nc;

#pragma unroll
    for (int q = 0; q < 4; ++q) {
        const int n = ng * 64 + q * 16 + nlo;
#pragma unroll
        for (int r = 0; r < 8; ++r) {
            const int p = pb * 16 + mhi + r;
            const float v = acc[q][r];
            if (useAtomic) atomicAdd(outBase + (size_t)p * Nc + n, v);
            else           outBase[(size_t)p * Nc + n] = v;
        }
    }
}

// -------------------------------------------------------------------------
__global__ __launch_bounds__(256) void reduce_partials(
    const float* __restrict__ part, float* __restrict__ out, size_t M)
{
    const size_t i = (size_t)blockIdx.x * blockDim.x + threadIdx.x;
    if (i < M) {
        float v = 0.f;
#pragma unroll
        for (int s = 0; s < SPLIT; ++s) v += part[(size_t)s * M + i];
        out[i] = v;
    }
}

__global__ __launch_bounds__(256) void zero_out(float* __restrict__ out, size_t M)
{
    const size_t i = (size_t)blockIdx.x * blockDim.x + threadIdx.x;
    if (i < M) out[i] = 0.f;
}

// -------------------------------------------------------------------------
extern "C" void kernel_launch(void* const* d_in, const int* in_sizes, int n_in,
                              void* d_out, int out_size, void* d_ws, size_t ws_size,
                              hipStream_t stream)
{
    const float* X = (const float*)d_in[0];
    const float* A = (const float*)d_in[1];
    const float* B = (const float*)d_in[2];
    // d_in[3] = C is unused by the reference forward (dead code there too)
    float* out = (float*)d_out;

    const size_t M      = (size_t)BATCHc * Hc * Pc * Nc;      // 1,048,576
    const size_t needWs = (size_t)SPLIT * M * sizeof(float);  // 16 MB

    const dim3 grid(SPLIT * BATCHc * Hc);   // 512 workgroups
    const dim3 block(256);                  // 8 waves (wave32)

    if (ws_size >= needWs) {
        float* part = (float*)d_ws;
        mamba_final_state_partial<<<grid, block, 0, stream>>>(X, A, B, part, 0);
        reduce_partials<<<dim3((unsigned)((M + 255) / 256)), block, 0, stream>>>(part, out, M);
    } else {
        zero_out<<<dim3((unsigned)((M + 255) / 256)), block, 0, stream>>>(out, M);
        mamba_final_state_partial<<<grid, block, 0, stream>>>(X, A, B, out, 1);
    }
}